// GNN_52295521796842
// MI455X (gfx1250) — compile-verified
//
#include <hip/hip_runtime.h>
#include <hip/hip_bf16.h>

typedef float v2f __attribute__((ext_vector_type(2)));
typedef float v8f __attribute__((ext_vector_type(8)));

#define FIN    512
#define HID    16
#define NCLS   10

// ---------------------------------------------------------------------------
// Init: deg=1 (self loop), agg1=0, out=0, build zero-padded W2 (16x16)
// ---------------------------------------------------------------------------
__global__ void k_init(float* __restrict__ agg1, float* __restrict__ deg,
                       float* __restrict__ out, float* __restrict__ w2pad,
                       const float* __restrict__ W2, int N) {
    int i = blockIdx.x * blockDim.x + threadIdx.x;
    long long tot = (long long)N * HID;
    if (i < tot) agg1[i] = 0.0f;
    if (i < N) deg[i] = 1.0f;                       // self-loop contribution
    if (i < N * NCLS) out[i] = 0.0f;
    if (i < HID * HID) {                            // W2 padded 16x10 -> 16x16
        int r = i >> 4, c = i & 15;
        w2pad[i] = (c < NCLS) ? W2[r * NCLS + c] : 0.0f;
    }
}

// ---------------------------------------------------------------------------
// Degree: deg[dst[e]] += 1
// ---------------------------------------------------------------------------
__global__ void k_degree(const int* __restrict__ dst, float* __restrict__ deg, int E) {
    int e = blockIdx.x * blockDim.x + threadIdx.x;
    if (e < E) atomicAdd(&deg[dst[e]], 1.0f);
}

// dinv = rsqrt(deg), in place (deg >= 1 always)
__global__ void k_dinv(float* __restrict__ deg, int N) {
    int n = blockIdx.x * blockDim.x + threadIdx.x;
    if (n < N) deg[n] = rsqrtf(deg[n]);
}

// ---------------------------------------------------------------------------
// GEMM1: H1[N,16] = X[N,512] @ W1[512,16]  (fp32 WMMA 16x16x4)
// 8 waves / block, each wave owns one 16-row tile. W1 staged in LDS (32 KB).
// ---------------------------------------------------------------------------
__global__ void __launch_bounds__(256)
k_gemm1(const float* __restrict__ X, const float* __restrict__ W1,
        float* __restrict__ H1, int N) {
    __shared__ float wlds[FIN * HID];               // 32 KB
    const int tid = threadIdx.x;
    for (int i = tid; i < FIN * HID; i += 256) wlds[i] = W1[i];
    __syncthreads();

    const int wave = tid >> 5;
    const int lane = tid & 31;
    const int half = lane >> 4;                     // 0: K pair {0,1}, 1: {2,3}
    const int l16  = lane & 15;
    const int tileRow = (blockIdx.x * 8 + wave) * 16;
    const int row  = tileRow + l16;                 // A-matrix M = lane%16
    const float mask = (row < N) ? 1.0f : 0.0f;     // zero OOB rows w/o branching
    const int rc   = (row < N) ? row : (N - 1);
    const float* __restrict__ xrow = X + (size_t)rc * FIN;

    v8f c = {0.f, 0.f, 0.f, 0.f, 0.f, 0.f, 0.f, 0.f};
    for (int k0 = 0; k0 < FIN; k0 += 4) {
        const int ka = k0 + (half << 1);
        v2f a, b;
        a.x = xrow[ka]     * mask;                  // A: M=l16, K=ka
        a.y = xrow[ka + 1] * mask;                  // A: M=l16, K=ka+1
        b.x = wlds[ka * HID + l16];                 // B: K=ka,   N=l16
        b.y = wlds[(ka + 1) * HID + l16];           // B: K=ka+1, N=l16
        c = __builtin_amdgcn_wmma_f32_16x16x4_f32(
                false, a, false, b, (short)0, c, false, false);
    }
    #pragma unroll
    for (int v = 0; v < 8; ++v) {                   // D: VGPR v -> rows v, v+8
        int r = tileRow + v + half * 8;
        if (r < N) H1[(size_t)r * HID + l16] = c[v];
    }
}

// ---------------------------------------------------------------------------
// Edge aggregation, layer 1: agg1[dst] += H1[src] * dinv[src]*dinv[dst]
// 16 lanes per edge -> contiguous 64B gathers and atomic segments
// ---------------------------------------------------------------------------
__global__ void k_agg1(const int* __restrict__ src, const int* __restrict__ dst,
                       const float* __restrict__ dinv, const float* __restrict__ H1,
                       float* __restrict__ agg1, int E) {
    long long t = (long long)blockIdx.x * blockDim.x + threadIdx.x;
    long long e = t >> 4;
    int f = (int)(t & 15);
    if (e >= E) return;
    int s = src[e], d = dst[e];
    float w = dinv[s] * dinv[d];
    atomicAdd(&agg1[(size_t)d * HID + f], H1[(size_t)s * HID + f] * w);
}

// Finalize layer 1: Hrelu = relu(agg1 + H1*dinv^2 + b1), written into H1 in place
__global__ void k_fin1(float* __restrict__ H1, const float* __restrict__ agg1,
                       const float* __restrict__ dinv, const float* __restrict__ b1,
                       int N) {
    long long i = (long long)blockIdx.x * blockDim.x + threadIdx.x;
    if (i >= (long long)N * HID) return;
    int n = (int)(i >> 4), f = (int)(i & 15);
    float di = dinv[n];
    float v = agg1[i] + H1[i] * di * di + b1[f];
    H1[i] = fmaxf(v, 0.0f);
}

// ---------------------------------------------------------------------------
// GEMM2: H2[N,16] = Hrelu[N,16] @ W2pad[16,16]  (fp32 WMMA, K=16 -> 4 steps)
// ---------------------------------------------------------------------------
__global__ void __launch_bounds__(256)
k_gemm2(const float* __restrict__ Hrelu, const float* __restrict__ W2pad,
        float* __restrict__ H2, int N) {
    __shared__ float wlds[HID * HID];
    if (threadIdx.x < HID * HID) wlds[threadIdx.x] = W2pad[threadIdx.x];
    __syncthreads();

    const int tid = threadIdx.x;
    const int wave = tid >> 5;
    const int lane = tid & 31;
    const int half = lane >> 4;
    const int l16  = lane & 15;
    const int tileRow = (blockIdx.x * 8 + wave) * 16;
    const int row  = tileRow + l16;
    const float mask = (row < N) ? 1.0f : 0.0f;
    const int rc   = (row < N) ? row : (N - 1);
    const float* __restrict__ hrow = Hrelu + (size_t)rc * HID;

    v8f c = {0.f, 0.f, 0.f, 0.f, 0.f, 0.f, 0.f, 0.f};
    #pragma unroll
    for (int k0 = 0; k0 < HID; k0 += 4) {
        const int ka = k0 + (half << 1);
        v2f a, b;
        a.x = hrow[ka]     * mask;
        a.y = hrow[ka + 1] * mask;
        b.x = wlds[ka * HID + l16];
        b.y = wlds[(ka + 1) * HID + l16];
        c = __builtin_amdgcn_wmma_f32_16x16x4_f32(
                false, a, false, b, (short)0, c, false, false);
    }
    #pragma unroll
    for (int v = 0; v < 8; ++v) {
        int r = tileRow + v + half * 8;
        if (r < N) H2[(size_t)r * HID + l16] = c[v];
    }
}

// ---------------------------------------------------------------------------
// Edge aggregation, layer 2: out[dst,c] += H2[src,c] * norm   (c < 10)
// ---------------------------------------------------------------------------
__global__ void k_agg2(const int* __restrict__ src, const int* __restrict__ dst,
                       const float* __restrict__ dinv, const float* __restrict__ H2,
                       float* __restrict__ out, int E) {
    long long t = (long long)blockIdx.x * blockDim.x + threadIdx.x;
    long long e = t >> 4;
    int f = (int)(t & 15);
    if (e >= E || f >= NCLS) return;
    int s = src[e], d = dst[e];
    float w = dinv[s] * dinv[d];
    atomicAdd(&out[(size_t)d * NCLS + f], H2[(size_t)s * HID + f] * w);
}

// Finalize layer 2: out += H2*dinv^2 + b2
__global__ void k_fin2(float* __restrict__ out, const float* __restrict__ H2,
                       const float* __restrict__ dinv, const float* __restrict__ b2,
                       int N) {
    long long i = (long long)blockIdx.x * blockDim.x + threadIdx.x;
    if (i >= (long long)N * NCLS) return;
    int n = (int)(i / NCLS), c = (int)(i % NCLS);
    float di = dinv[n];
    out[i] += H2[(size_t)n * HID + c] * di * di + b2[c];
}

// ---------------------------------------------------------------------------
extern "C" void kernel_launch(void* const* d_in, const int* in_sizes, int n_in,
                              void* d_out, int out_size, void* d_ws, size_t ws_size,
                              hipStream_t stream) {
    const float* X  = (const float*)d_in[0];
    const int*   EI = (const int*)d_in[1];        // int32 edge index [2,E]
    const float* W1 = (const float*)d_in[2];
    const float* b1 = (const float*)d_in[3];
    const float* W2 = (const float*)d_in[4];
    const float* b2 = (const float*)d_in[5];
    float* out = (float*)d_out;

    const int N = in_sizes[0] / FIN;              // 100000
    const int E = in_sizes[1] / 2;                // 3200000
    const int* src = EI;
    const int* dst = EI + E;

    // workspace layout (floats)
    float* H1    = (float*)d_ws;                  // N*16
    float* agg1  = H1   + (size_t)N * HID;        // N*16
    float* H2    = agg1 + (size_t)N * HID;        // N*16 (cols 10..15 are zero)
    float* dinv  = H2   + (size_t)N * HID;        // N  (deg -> rsqrt in place)
    float* w2pad = dinv + N;                      // 256

    const int B = 256;
    const long long nh = (long long)N * HID;
    const long long et = (long long)E * HID;      // edge * 16 lanes

    k_init  <<<(int)((nh + B - 1) / B), B, 0, stream>>>(agg1, dinv, out, w2pad, W2, N);
    k_degree<<<(E + B - 1) / B, B, 0, stream>>>(dst, dinv, E);
    k_dinv  <<<(N + B - 1) / B, B, 0, stream>>>(dinv, N);
    k_gemm1 <<<(N + 127) / 128, B, 0, stream>>>(X, W1, H1, N);
    k_agg1  <<<(int)((et + B - 1) / B), B, 0, stream>>>(src, dst, dinv, H1, agg1, E);
    k_fin1  <<<(int)((nh + B - 1) / B), B, 0, stream>>>(H1, agg1, dinv, b1, N);
    k_gemm2 <<<(N + 127) / 128, B, 0, stream>>>(H1, w2pad, H2, N);
    k_agg2  <<<(int)((et + B - 1) / B), B, 0, stream>>>(src, dst, dinv, H2, out, E);
    k_fin2  <<<(int)(((long long)N * NCLS + B - 1) / B), B, 0, stream>>>(out, H2, dinv, b2, N);
}